// DDNLoss_81295140978757
// MI455X (gfx1250) — compile-verified
//
#include <hip/hip_runtime.h>
#include <math.h>
#include <stdint.h>

// ---- problem constants (match reference) ----
#define B_   8
#define C_   81
#define H_   192
#define W_   640
#define HI_  384
#define WI_  1280
#define NPI_ 16
#define HW_  (H_ * W_)                    // 122880

#define BLOCK_THREADS  256
#define WAVES          8                  // 256 / wave32
#define PIX_PER_LANE   4
#define WAVE_PIX       128                // 32 lanes * 4 pixels
#define PIX_PER_BLOCK  (WAVES * WAVE_PIX) // 1024
#define NBLOCKS        ((B_ * HW_) / PIX_PER_BLOCK)   // 960

// ---- TDM streaming config ----
#define D_CH        9                     // channels per TDM chunk (81 = 9*9)
#define NCHUNK      (C_ / D_CH)           // 9
#define SLOT_FLOATS (WAVE_PIX * D_CH)     // 1152 floats = 4608 B per wave per buffer
#define SLOT_BYTES  (SLOT_FLOATS * 4)

typedef uint32_t u32x4 __attribute__((ext_vector_type(4)));
typedef uint32_t u32x8 __attribute__((ext_vector_type(8)));

// asm waits with "memory" clobber: hard ordering fence for the ds_loads that
// consume freshly DMA'd LDS (TENSORcnt retires in order per ISA 15.5).
#define WAIT_TCNT_0() asm volatile("s_wait_tensorcnt 0x0" ::: "memory")
#define WAIT_TCNT_1() asm volatile("s_wait_tensorcnt 0x1" ::: "memory")

// Issue one TDM 2D tile load: WAVE_PIX contiguous floats x D_CH channels,
// channel stride HW_, into LDS at lds_addr. Descriptor per CDNA5 ISA ch.8.
__device__ __forceinline__ void tdm_load_chunk(uint64_t gaddr, uint32_t lds_addr)
{
    u32x4 g0;
    g0[0] = 1u;                                        // count=1 (valid user D#)
    g0[1] = lds_addr;                                  // LDS byte address
    g0[2] = (uint32_t)gaddr;                           // global_addr[31:0]
    g0[3] = ((uint32_t)(gaddr >> 32) & 0x01FFFFFFu)    // global_addr[56:32]
          | (2u << 30);                                // type = 2 ("image")

    u32x8 g1;
    g1[0] = (2u << 16);                                // wg_mask=0, data_size=4B
    g1[1] = ((uint32_t)WAVE_PIX & 0xFFFFu) << 16;      // tensor_dim0[15:0]
    g1[2] = (((uint32_t)WAVE_PIX >> 16) & 0xFFFFu)     // tensor_dim0[31:16]
          | (((uint32_t)D_CH & 0xFFFFu) << 16);        // tensor_dim1[15:0]
    g1[3] = (((uint32_t)D_CH >> 16) & 0xFFFFu)         // tensor_dim1[31:16]
          | ((uint32_t)WAVE_PIX << 16);                // tile_dim0 = 128
    g1[4] = (uint32_t)D_CH;                            // tile_dim1 = 9, tile_dim2 = 0
    g1[5] = (uint32_t)HW_;                             // tensor_dim0_stride[31:0]
    g1[6] = 0u;                                        // stride0[47:32] | stride1[15:0]
    g1[7] = 0u;                                        // stride1[47:16]

    // 2-group form: VADDR2/VADDR3 NULL -> up to 2D tensor
    asm volatile("tensor_load_to_lds %0, %1" :: "s"(g0), "s"(g1) : "memory");
}

// Stage 1: per-block partial weighted focal loss (pre-scaled by 1/num_pixels)
__global__ __launch_bounds__(BLOCK_THREADS)
void ddn_partial(const float* __restrict__ logits,     // [B,C,H,W]
                 const float* __restrict__ boxes,      // [B*NPI, 4]
                 const float* __restrict__ depths,     // [B*NPI]
                 const int*   __restrict__ instances,  // [B,HI,WI]
                 const int*   __restrict__ maskids,    // [B,NPI]
                 float*       __restrict__ partial)    // [NBLOCKS]
{
    const int tid  = threadIdx.x;
    const int lane = tid & 31;
    const int wave = __builtin_amdgcn_readfirstlane(tid >> 5);

    const int pix_base = blockIdx.x * PIX_PER_BLOCK;   // global pixel index
    const int b        = pix_base / HW_;               // uniform per block
    const int q0w      = (pix_base - b * HW_) + wave * WAVE_PIX; // wave's 1st pixel
    const int q0       = q0w + lane * PIX_PER_LANE;

    // ---------- per-pixel preamble: target bin + fg weight ----------
    int   targ[PIX_PER_LANE];
    float wgt [PIX_PER_LANE];

    const float bin_size = 2.0f * (60.0f - 0.001f) / (80.0f * 81.0f);
    const float inv_bin8 = 8.0f / bin_size;

    #pragma unroll
    for (int i = 0; i < PIX_PER_LANE; ++i) {
        const int q = q0 + i;
        const int y = q / W_;
        const int x = q - y * W_;

        // nearest resize (2x): src = 2*dst + 1
        const int id = instances[b * (HI_ * WI_) + (2 * y + 1) * WI_ + (2 * x + 1)];

        float dmin = INFINITY;
        #pragma unroll 1
        for (int p = 0; p < NPI_; ++p) {
            const int   mid = 1000 + maskids[b * NPI_ + p];
            const float d   = depths[b * NPI_ + p];
            if (mid == id) dmin = fminf(dmin, d);
        }
        const float dep = (dmin == INFINITY) ? 0.0f : dmin;

        const float idx  = -0.5f + 0.5f * sqrtf(1.0f + (dep - 0.001f) * inv_bin8);
        const bool  good = (idx >= 0.0f) && (idx <= 80.0f);   // NaN/Inf -> bad
        targ[i] = good ? (int)idx : 80;

        bool fg = false;
        #pragma unroll 1
        for (int p = 0; p < NPI_; ++p) {
            const float* bx = boxes + (b * NPI_ + p) * 4;
            const int u1 = (int)floorf(bx[0]);
            const int v1 = (int)floorf(bx[1]);
            const int u2 = (int)ceilf(bx[2]);
            const int v2 = (int)ceilf(bx[3]);
            fg = fg | ((x >= u1) & (x < u2) & (y >= v1) & (y < v2));
        }
        wgt[i] = fg ? 13.0f : 1.0f;
    }

    // ---------- TDM double-buffered LDS stream, online softmax ----------
    __shared__ __align__(16) float smem[2 * WAVES * SLOT_FLOATS];   // 73728 B

    const uint32_t lds_base = (uint32_t)(uintptr_t)(&smem[0]);      // LDS byte offset
    const uint32_t slot0    = lds_base + (uint32_t)(wave * SLOT_BYTES);
    const uint32_t slot1    = slot0 + (uint32_t)(WAVES * SLOT_BYTES);

    const uint64_t gbase = (uint64_t)(uintptr_t)logits
                         + 4ull * ((uint64_t)b * C_ * HW_ + (uint64_t)q0w);

    float m[PIX_PER_LANE]  = {-INFINITY, -INFINITY, -INFINITY, -INFINITY};
    float s[PIX_PER_LANE]  = {0.0f, 0.0f, 0.0f, 0.0f};
    float xt[PIX_PER_LANE] = {0.0f, 0.0f, 0.0f, 0.0f};

    tdm_load_chunk(gbase, slot0);                       // prologue: chunk 0
    for (int k = 0; k < NCHUNK; ++k) {
        if (k + 1 < NCHUNK) {
            const uint32_t nslot = ((k + 1) & 1) ? slot1 : slot0;
            tdm_load_chunk(gbase + 4ull * (uint64_t)(k + 1) * D_CH * HW_, nslot);
            WAIT_TCNT_1();                              // in-order: chunk k retired
        } else {
            WAIT_TCNT_0();
        }

        const float* sp = &smem[((k & 1) * WAVES + wave) * SLOT_FLOATS
                                + lane * PIX_PER_LANE];
        #pragma unroll
        for (int ch = 0; ch < D_CH; ++ch) {
            const int    c = k * D_CH + ch;
            const float4 v = *(const float4*)(sp + ch * WAVE_PIX); // ds_load_b128
            const float xs[4] = {v.x, v.y, v.z, v.w};
            #pragma unroll
            for (int i = 0; i < PIX_PER_LANE; ++i) {
                const float xi = xs[i];
                const float mo = m[i];
                const float mn = fmaxf(mo, xi);
                s[i]  = s[i] * __expf(mo - mn) + __expf(xi - mn);  // exp(-inf)=0 ok
                m[i]  = mn;
                xt[i] = (c == targ[i]) ? xi : xt[i];
            }
        }
    }

    // ---------- focal loss + fg/bg balancer ----------
    const float inv_np = 1.0f / (float)(B_ * HW_);
    float acc = 0.0f;
    #pragma unroll
    for (int i = 0; i < PIX_PER_LANE; ++i) {
        const float logpt = xt[i] - m[i] - __logf(s[i]);
        const float pt    = __expf(logpt);
        const float om    = 1.0f - pt;
        const float loss  = -0.25f * om * om * logpt;
        acc += loss * wgt[i];
    }
    acc *= inv_np;

    // ---------- deterministic block reduction ----------
    __shared__ float red[BLOCK_THREADS];
    red[tid] = acc;
    __syncthreads();
    #pragma unroll
    for (int off = BLOCK_THREADS / 2; off > 0; off >>= 1) {
        if (tid < off) red[tid] += red[tid + off];
        __syncthreads();
    }
    if (tid == 0) partial[blockIdx.x] = red[0];
}

// Stage 2: fixed-order reduction of the 960 partials -> scalar loss
__global__ __launch_bounds__(BLOCK_THREADS)
void ddn_final(const float* __restrict__ partial, float* __restrict__ out)
{
    __shared__ float red[BLOCK_THREADS];
    float acc = 0.0f;
    for (int i = threadIdx.x; i < NBLOCKS; i += BLOCK_THREADS)
        acc += partial[i];
    red[threadIdx.x] = acc;
    __syncthreads();
    #pragma unroll
    for (int off = BLOCK_THREADS / 2; off > 0; off >>= 1) {
        if (threadIdx.x < off) red[threadIdx.x] += red[threadIdx.x + off];
        __syncthreads();
    }
    if (threadIdx.x == 0) out[0] = red[0];
}

extern "C" void kernel_launch(void* const* d_in, const int* in_sizes, int n_in,
                              void* d_out, int out_size, void* d_ws, size_t ws_size,
                              hipStream_t stream)
{
    (void)in_sizes; (void)n_in; (void)out_size; (void)ws_size;
    const float* logits    = (const float*)d_in[0];  // depth_logits    [B,C,H,W] f32
    const float* boxes     = (const float*)d_in[1];  // gt_boxes2d      [B*NPI,4] f32
    const float* depths    = (const float*)d_in[2];  // gt_center_depth [B*NPI]   f32
    const int*   instances = (const int*)  d_in[3];  // instances       [B,HI,WI] i32
    const int*   maskids   = (const int*)  d_in[4];  // mask_instances  [B,NPI]   i32
    float*       out       = (float*)d_out;
    float*       partials  = (float*)d_ws;           // NBLOCKS floats of scratch

    ddn_partial<<<NBLOCKS, BLOCK_THREADS, 0, stream>>>(
        logits, boxes, depths, instances, maskids, partials);
    ddn_final<<<1, BLOCK_THREADS, 0, stream>>>(partials, out);
}